// GraphConvLayer_65481071397258
// MI455X (gfx1250) — compile-verified
//
#include <hip/hip_runtime.h>

// GraphConv: out[n] = W[type[n]] @ (A @ x)[n]
// N=16384, IN=OUT=128, T=8.
// Roofline: A is 1 GiB fp32 read once -> ~46us floor @ 23.3 TB/s; required
// compute is ~1.5 PFLOP/s, so stay fp32-exact (v_wmma_f32_16x16x4_f32)
// and optimize the HBM stream:
//   - A chunks: GLOBAL_LOAD_ASYNC_TO_LDS_B128, double-buffered (ASYNCcnt),
//     prefetch of chunk k+1 overlapped with WMMA on chunk k.
//   - x chunks (L2-resident): global->VGPR->LDS transpose so every WMMA
//     B-fragment is a single aligned ds_load_b64 (no VGPR marshaling).

typedef __attribute__((ext_vector_type(2))) float v2f;
typedef __attribute__((ext_vector_type(8))) float v8f;

#define N_NODES    16384
#define IN_DIM     128
#define OUT_DIM    128
#define NUM_TYPES  8
#define KC         32            // K-chunk staged through LDS
#define MBLK       64            // A-rows per workgroup (4 waves x 16 rows)
#define A_STRIDE   36            // KC+4 pad: 144B rows -> 16B-aligned b128 async
                                 // segments, 8B-aligned b64 frags, 16 banks hit
#define X_STRIDE   36            // x^T tile stride, same alignment properties
#define AGG_STRIDE 130           // IN_DIM + 2 pad for stage-2 agg tile

#define SA_ELEMS   (MBLK * A_STRIDE)       // 2304
#define SXT_ELEMS  (IN_DIM * X_STRIDE)     // 4608
#define SMEM_ELEMS (2 * SA_ELEMS + 2 * SXT_ELEMS)   // 13824 floats = 55296 B

// ---- CDNA5 async memory->LDS copy (16B per lane, tracked by ASYNCcnt) ----
__device__ __forceinline__ void async_load_b128(const float* g, const float* lds) {
    asm volatile("global_load_async_to_lds_b128 %0, %1, off"
                 :: "v"((unsigned int)(unsigned long long)lds),
                    "v"((unsigned long long)g)
                 : "memory");
}

__device__ __forceinline__ void wait_async0() {
#if __has_builtin(__builtin_amdgcn_s_wait_asynccnt)
    __builtin_amdgcn_s_wait_asynccnt(0);
#else
    asm volatile("s_wait_asynccnt 0x0" ::: "memory");
#endif
}

__global__ __launch_bounds__(128)
void graphconv_fused(const float* __restrict__ x,
                     const int*   __restrict__ types,
                     const float* __restrict__ adj,
                     const float* __restrict__ wgt,
                     float*       __restrict__ out)
{
    __shared__ float smem[SMEM_ELEMS];

    const int tid  = threadIdx.x;
    const int w    = tid >> 5;                 // wave 0..3, owns 16 rows
    const int lane = tid & 31;
    const int lo   = lane & 15;                // M / N index within fragment
    const int hi   = lane >> 4;                // K half-select (16x4 f32 layout)
    const int rowBase = blockIdx.x * MBLK;

    // A chunk: MBLK x KC = 512 x 16B segments, 4 async copies per thread.
    auto issueA = [&](int k0, int buf) {
        float* sA = smem + buf * SA_ELEMS;
#pragma unroll
        for (int i = 0; i < 4; ++i) {
            int idx = i * 128 + tid;           // 0..511
            int r   = idx >> 3;
            int c4  = (idx & 7) << 2;
            async_load_b128(adj + (size_t)(rowBase + r) * N_NODES + k0 + c4,
                            sA + r * A_STRIDE + c4);
        }
    };
    // x chunk: KC x IN_DIM as 8 float4 per thread (issued early, consumed late).
    auto loadX = [&](int k0, float4* v) {
#pragma unroll
        for (int i = 0; i < 8; ++i) {
            int idx = i * 128 + tid;           // 0..1023
            int k   = idx >> 5;
            int n4  = (idx & 31) << 2;
            v[i] = *(const float4*)(x + (size_t)(k0 + k) * IN_DIM + n4);
        }
    };
    auto storeXt = [&](const float4* v, int buf) {
        float* sXt = smem + 2 * SA_ELEMS + buf * SXT_ELEMS;
#pragma unroll
        for (int i = 0; i < 8; ++i) {
            int idx = i * 128 + tid;
            int k   = idx >> 5;
            int n4  = (idx & 31) << 2;
            sXt[(n4 + 0) * X_STRIDE + k] = v[i].x;
            sXt[(n4 + 1) * X_STRIDE + k] = v[i].y;
            sXt[(n4 + 2) * X_STRIDE + k] = v[i].z;
            sXt[(n4 + 3) * X_STRIDE + k] = v[i].w;
        }
    };

    v8f acc[8];
#pragma unroll
    for (int nt = 0; nt < 8; ++nt) acc[nt] = (v8f){0,0,0,0,0,0,0,0};

    // ---------------- Stage 1: agg = A @ x (double-buffered) ----------------
    {
        issueA(0, 0);
        float4 xv[8];
        loadX(0, xv);
        storeXt(xv, 0);
        wait_async0();
    }
    __syncthreads();

    for (int k0 = 0, buf = 0; k0 < N_NODES; k0 += KC, buf ^= 1) {
        const int nk0 = k0 + KC;
        float4 xv[8];
        if (nk0 < N_NODES) {                   // prefetch overlaps compute below
            issueA(nk0, buf ^ 1);
            loadX(nk0, xv);
        }

        const float* sA  = smem + buf * SA_ELEMS;
        const float* sXt = smem + 2 * SA_ELEMS + buf * SXT_ELEMS;
#pragma unroll
        for (int kk = 0; kk < KC / 4; ++kk) {
            const int kp = kk * 4 + 2 * hi;
            const v2f a = *(const v2f*)(sA + (w * 16 + lo) * A_STRIDE + kp);
#pragma unroll
            for (int nt = 0; nt < 8; ++nt) {
                const v2f b = *(const v2f*)(sXt + (nt * 16 + lo) * X_STRIDE + kp);
                acc[nt] = __builtin_amdgcn_wmma_f32_16x16x4_f32(
                    false, a, false, b, (short)0, acc[nt], false, false);
            }
        }

        if (nk0 < N_NODES) {
            storeXt(xv, buf ^ 1);              // x regs -> LDS (loads done by now)
            wait_async0();                     // A chunk resident in LDS
        }
        __syncthreads();
    }

    // ---- spill agg tile (64 x 128) to LDS; C layout: VGPR i -> M=i+8*hi, N=lo ----
    float* sAgg = smem;                        // aliases the K-chunk buffers
#pragma unroll
    for (int nt = 0; nt < 8; ++nt)
#pragma unroll
        for (int i = 0; i < 8; ++i)
            sAgg[(w * 16 + i + 8 * hi) * AGG_STRIDE + nt * 16 + lo] = acc[nt][i];
    __syncthreads();

    // ---------------- Stage 2: typed projection ----------------
    // out[n] = W[type[n]] @ agg[n]; exact via per-type row masking, summed over t.
    const int myType = types[rowBase + w * 16 + lo];
    const v2f z2 = (v2f){0.f, 0.f};
#pragma unroll 1
    for (int ot = 0; ot < 8; ++ot) {
        v8f c = (v8f){0,0,0,0,0,0,0,0};
        for (int kk = 0; kk < IN_DIM / 4; ++kk) {
            const int kp = kk * 4 + 2 * hi;
            const v2f araw = *(const v2f*)(sAgg + (w * 16 + lo) * AGG_STRIDE + kp);
#pragma unroll
            for (int t = 0; t < NUM_TYPES; ++t) {
                const v2f am = (myType == t) ? araw : z2;   // mask rows not of type t
                // B[d][o] = W[t][o][d]; 2 consecutive d -> 8B load, L2-resident
                const v2f b = *(const v2f*)(wgt + ((size_t)t * OUT_DIM + ot * 16 + lo) * IN_DIM + kp);
                c = __builtin_amdgcn_wmma_f32_16x16x4_f32(
                    false, am, false, b, (short)0, c, false, false);
            }
        }
#pragma unroll
        for (int i = 0; i < 8; ++i)
            out[(size_t)(rowBase + w * 16 + i + 8 * hi) * OUT_DIM + ot * 16 + lo] = c[i];
    }
}

extern "C" void kernel_launch(void* const* d_in, const int* in_sizes, int n_in,
                              void* d_out, int out_size, void* d_ws, size_t ws_size,
                              hipStream_t stream) {
    const float* x   = (const float*)d_in[0];
    const int*   nt  = (const int*)d_in[1];
    const float* adj = (const float*)d_in[2];
    const float* wgt = (const float*)d_in[3];
    float*       out = (float*)d_out;

    dim3 grid(N_NODES / MBLK);
    dim3 block(128);
    hipLaunchKernelGGL(graphconv_fused, grid, block, 0, stream, x, nt, adj, wgt, out);
}